// GATLayer_58291296141986
// MI455X (gfx1250) — compile-verified
//
#include <hip/hip_runtime.h>
#include <hip/hip_bf16.h>

typedef __attribute__((ext_vector_type(2))) float v2f;
typedef __attribute__((ext_vector_type(8))) float v8f;

#define BATCH    32
#define NNODE    10
#define IN_DIM   65536
#define OUT_DIM  64
#define HEADS    8
#define ROWS     (BATCH * NNODE)   // 320
#define ROW_TILES 20               // ROWS / 16
#define KC        32               // split-K chunks
#define KLEN      (IN_DIM / KC)    // 2048

// ---------------------------------------------------------------------------
// GEMM: Y = X(320 x 65536) @ W^T(65536 x 64), split-K partials in ws.
// 1 wave per (rowTile, kChunk); 4 column tiles per wave share the A fragment.
// V_WMMA_F32_16X16X4_F32, fp32 end-to-end (matches reference precision).
// ---------------------------------------------------------------------------
__global__ __launch_bounds__(32)
void gat_gemm_wmma(const float* __restrict__ x,
                   const float* __restrict__ W,
                   float* __restrict__ partial) {
    const int rowTile = blockIdx.x % ROW_TILES;
    const int kc      = blockIdx.x / ROW_TILES;
    const int lane    = threadIdx.x & 31;
    const int mrow    = lane & 15;          // A: row / B,C: column within tile
    const int khalf   = (lane >> 4) << 1;   // lanes 16..31 hold K+2, K+3
    const int kstart  = kc * KLEN;

    const float* xp  = x + (size_t)(rowTile * 16 + mrow) * IN_DIM + kstart + khalf;
    const float* wp0 = W + (size_t)( 0 + mrow) * IN_DIM + kstart + khalf;
    const float* wp1 = W + (size_t)(16 + mrow) * IN_DIM + kstart + khalf;
    const float* wp2 = W + (size_t)(32 + mrow) * IN_DIM + kstart + khalf;
    const float* wp3 = W + (size_t)(48 + mrow) * IN_DIM + kstart + khalf;

    v8f c0 = {0.f,0.f,0.f,0.f,0.f,0.f,0.f,0.f};
    v8f c1 = c0, c2 = c0, c3 = c0;

    for (int k = 0; k < KLEN; k += 32) {
        // prefetch the x stream ~2KB ahead (speculative, dropped if OOB)
        if (k + 512 < KLEN) __builtin_prefetch(xp + k + 512, 0, 0);
        #pragma unroll
        for (int kk = 0; kk < 32; kk += 4) {
            v2f a  = *(const v2f*)(xp  + k + kk);
            v2f b0 = *(const v2f*)(wp0 + k + kk);
            v2f b1 = *(const v2f*)(wp1 + k + kk);
            v2f b2 = *(const v2f*)(wp2 + k + kk);
            v2f b3 = *(const v2f*)(wp3 + k + kk);
            c0 = __builtin_amdgcn_wmma_f32_16x16x4_f32(false, a, false, b0, (short)0, c0, false, false);
            c1 = __builtin_amdgcn_wmma_f32_16x16x4_f32(false, a, false, b1, (short)0, c1, false, false);
            c2 = __builtin_amdgcn_wmma_f32_16x16x4_f32(false, a, false, b2, (short)0, c2, false, false);
            c3 = __builtin_amdgcn_wmma_f32_16x16x4_f32(false, a, false, b3, (short)0, c3, false, false);
        }
    }

    // C layout: VGPR i, lanes 0-15 -> (M=i,   N=lane),
    //                   lanes 16-31-> (M=i+8, N=lane-16)
    float* pt = partial + ((size_t)kc * ROWS + rowTile * 16) * OUT_DIM;
    const int rhalf = (lane >> 4) * 8;
    #pragma unroll
    for (int i = 0; i < 8; ++i) {
        const size_t r = (size_t)(i + rhalf) * OUT_DIM + mrow;
        pt[r +  0] = c0[i];
        pt[r + 16] = c1[i];
        pt[r + 32] = c2[i];
        pt[r + 48] = c3[i];
    }
}

// ---------------------------------------------------------------------------
// Epilogue: s[row,h] = sum over KC chunks and 8 head columns (fixed order ->
// deterministic), then out[row, h*10 + m] = relu(s * adj[row, m]).
// ---------------------------------------------------------------------------
__global__ __launch_bounds__(256)
void gat_epilogue(const float* __restrict__ partial,
                  const float* __restrict__ adj,
                  float* __restrict__ out) {
    const int tid = blockIdx.x * blockDim.x + threadIdx.x;
    if (tid >= ROWS * HEADS) return;
    const int row = tid >> 3;
    const int h   = tid & 7;

    float s = 0.f;
    for (int kc = 0; kc < KC; ++kc) {
        const float* p = partial + ((size_t)kc * ROWS + row) * OUT_DIM + h * 8;
        #pragma unroll
        for (int d = 0; d < 8; ++d) s += p[d];
    }

    const float* arow = adj + (size_t)row * NNODE;
    float*       orow = out + (size_t)row * (HEADS * NNODE) + h * NNODE;
    #pragma unroll
    for (int m = 0; m < NNODE; ++m) {
        const float v = s * arow[m];
        orow[m] = v > 0.f ? v : 0.f;
    }
}

extern "C" void kernel_launch(void* const* d_in, const int* in_sizes, int n_in,
                              void* d_out, int out_size, void* d_ws, size_t ws_size,
                              hipStream_t stream) {
    const float* x   = (const float*)d_in[0];  // (32, 10, 65536) f32
    const float* adj = (const float*)d_in[1];  // (32, 10, 10)    f32
    const float* W   = (const float*)d_in[2];  // (64, 65536)     f32
    float*       out = (float*)d_out;          // (32, 10, 80)    f32

    float* partial = (float*)d_ws;             // KC*320*64 f32 = 2.62 MB

    gat_gemm_wmma<<<ROW_TILES * KC, 32, 0, stream>>>(x, W, partial);
    gat_epilogue<<<(ROWS * HEADS + 255) / 256, 256, 0, stream>>>(partial, adj, out);
}